// SelfAttentionLayer_55413668053553
// MI455X (gfx1250) — compile-verified
//
#include <hip/hip_runtime.h>

// ---------------------------------------------------------------------------
// Shapes (fixed by reference): N=1, T=8, D=64, C=32, H=W=56
//   B  = D*C = 2048 independent attention problems
//   S  = H*W = 3136 (sequence/feature length per row)
//   Row t of batch b starts at float offset t*TSTR + b*S (contiguous in s).
// Strategy: 1 workgroup (256 thr = 8 waves) per batch.
//   1) GLOBAL_LOAD_ASYNC_TO_LDS_B128 stages the 8x3136 tile (100 KB) -> LDS
//      (single HBM pass; ASYNCcnt + s_wait_asynccnt)
//   2) Gram matrix row via V_WMMA_F32_16X16X4_F32, 8 waves split K,
//      software-pipelined ds_load_b64 (uniform 98-trip loop, no exec masking)
//   3) 8-wide softmax: lanes 0..7 + width-8 shuffle butterfly
//   4) weighted sum from LDS, fused gamma*out + row0, nontemporal b128 stores
// ---------------------------------------------------------------------------

#define T_    8
#define S_    3136
#define PAD_  3140          // 3136 + 4: breaks LDS bank-conflict (3136 % 64 == 0)
#define TSTR_ 6422528       // D*C*H*W floats between consecutive t planes
#define S4_   784           // S_/4 float4 chunks per row
#define NSTEP_ 98           // 784 k-steps / 8 waves, uniform for every wave

typedef __attribute__((ext_vector_type(2))) float v2f;
typedef __attribute__((ext_vector_type(4))) float v4f;
typedef __attribute__((ext_vector_type(8))) float v8f;

// gcc-style int4 + address-space-qualified pointers matching the async builtin
typedef int v4i __attribute__((vector_size(16)));
typedef __attribute__((address_space(1))) v4i* g_v4i_ptr;   // global
typedef __attribute__((address_space(3))) v4i* l_v4i_ptr;   // LDS

#if defined(__gfx1250__) && \
    __has_builtin(__builtin_amdgcn_global_load_async_to_lds_b128) && \
    __has_builtin(__builtin_amdgcn_s_wait_asynccnt)
#define HAVE_ASYNC_LDS 1
#else
#define HAVE_ASYNC_LDS 0
#endif

__global__ __launch_bounds__(256) void
attn_t8_s3136_kernel(const float* __restrict__ x,
                     const float* __restrict__ gamma,
                     float* __restrict__ out)
{
    extern __shared__ float lds[];
    float* rows    = lds;                 // T_ * PAD_ floats (tile, row-padded)
    float* partial = lds + T_ * PAD_;     // 8 waves * 8 scores
    float* attn    = partial + 64;        // 8 softmax weights

    const int b    = blockIdx.x;
    const int tid  = threadIdx.x;
    const int lane = tid & 31;            // wave32
    const int wave = tid >> 5;

    const float* xb = x + (size_t)b * S_;

    // ---- 1) stage 8 rows into LDS (b128 async copies, coalesced) -----------
    for (int i = tid; i < T_ * S4_; i += 256) {
        const int t  = i / S4_;
        const int s4 = (i - t * S4_) * 4;
        const float* gp = xb + (size_t)t * TSTR_ + s4;
        float*       lp = rows + t * PAD_ + s4;
#if HAVE_ASYNC_LDS
        __builtin_amdgcn_global_load_async_to_lds_b128(
            (g_v4i_ptr)gp, (l_v4i_ptr)lp, /*offset=*/0, /*cpol=*/0);
#else
        *(v4f*)lp = *(const v4f*)gp;
#endif
    }
#if HAVE_ASYNC_LDS
    __builtin_amdgcn_s_wait_asynccnt(0);
#endif
    __syncthreads();

    // ---- 2) Gram partials: G = R * R^T over this wave's K-slice -------------
    // V_WMMA_F32_16X16X4_F32 A-matrix layout (32-bit, 16x4):
    //   lanes 0-15 : VGPR0 = A[M=lane][K=k+0], VGPR1 = K=k+1
    //   lanes16-31 : VGPR0 = A[M=lane-16][K=k+2], VGPR1 = K=k+3
    // B (4x16) mirrors with N across lanes; for a Gram matrix (B = A^T of the
    // same R) the per-lane register payloads coincide -> load once, use twice.
    // Trip count is uniformly 98 for all 8 waves -> scalar loop, EXEC stays
    // all-ones (WMMA requirement), and we prefetch the next ds_load_b64 so the
    // scheduler can overlap LDS latency with the WMMA.
    const int row  = lane & 7;                 // rows 8..15 duplicate 0..7 (unused region)
    const int koff = (lane < 16) ? 0 : 2;
    const float* p = rows + row * PAD_ + koff + wave * 4;  // step stride = 8*4 floats

    v8f acc = {};
    v2f ab = *(const v2f*)p;                   // prologue load
    p += 32;
    #pragma unroll 2
    for (int it = 0; it < NSTEP_ - 1; ++it) {
        v2f nxt = *(const v2f*)p;              // prefetch next chunk
        p += 32;
        acc = __builtin_amdgcn_wmma_f32_16x16x4_f32(
                  false, ab, false, ab, (short)0, acc, false, false);
        ab = nxt;
    }
    acc = __builtin_amdgcn_wmma_f32_16x16x4_f32(
              false, ab, false, ab, (short)0, acc, false, false);

    // acc[0] on lanes 0..15 is G[M=0][N=lane]; we need N = t = 0..7.
    if (lane < T_) partial[wave * T_ + lane] = acc[0];
    __syncthreads();

    // ---- 3) reduce partials + softmax over 8 scalars (lanes 0..7, wave 0) ---
    if (wave == 0 && lane < T_) {
        float v = 0.f;
        #pragma unroll
        for (int w = 0; w < 8; ++w) v += partial[w * T_ + lane];
        const float s = v * (1.0f / 56.0f);     // / sqrt(H*W)
        // width-8 butterfly: max then sum, all partners within lanes 0..7
        float m = s;
        m = fmaxf(m, __shfl_xor(m, 4, 8));
        m = fmaxf(m, __shfl_xor(m, 2, 8));
        m = fmaxf(m, __shfl_xor(m, 1, 8));
        const float e = __expf(s - m);
        float sum = e;
        sum += __shfl_xor(sum, 4, 8);
        sum += __shfl_xor(sum, 2, 8);
        sum += __shfl_xor(sum, 1, 8);
        attn[lane] = e / sum;
    }
    __syncthreads();

    // ---- 4) out = gamma * (attn . rows) + row0, streamed from LDS -----------
    const float g = *gamma;
    float a[T_];
    #pragma unroll
    for (int t = 0; t < T_; ++t) a[t] = attn[t];

    float* ob = out + (size_t)b * S_;
    for (int i = tid; i < S4_; i += 256) {
        const int s4 = i * 4;
        v4f o = {0.f, 0.f, 0.f, 0.f};
        #pragma unroll
        for (int t = 0; t < T_; ++t) {
            v4f r = *(const v4f*)(rows + t * PAD_ + s4);  // ds_load_b128
            o = o + a[t] * r;                             // v_fmac per component
        }
        v4f ref = *(const v4f*)(rows + s4);               // row 0
        v4f res = g * o + ref;
        __builtin_nontemporal_store(res, (v4f*)(ob + s4)); // NT global_store_b128
    }
}

extern "C" void kernel_launch(void* const* d_in, const int* in_sizes, int n_in,
                              void* d_out, int out_size, void* d_ws, size_t ws_size,
                              hipStream_t stream) {
    const float* x     = (const float*)d_in[0];
    const float* gamma = (const float*)d_in[1];
    float*       out   = (float*)d_out;

    const size_t lds_bytes = (size_t)(T_ * PAD_ + 64 + 8) * sizeof(float); // ~100.8 KB
    attn_t8_s3136_kernel<<<dim3(2048), dim3(256), lds_bytes, stream>>>(x, gamma, out);
}